// SelfAttention_24670292148457
// MI455X (gfx1250) — compile-verified
//
#include <hip/hip_runtime.h>
#include <hip/hip_bf16.h>

// MI455X / gfx1250 fused self-attention-over-heads (FP32 WMMA).
//   q,k,v = x@W.T + b ; 4x4 head-scores per token ; softmax ; y = p@v ; out = y@Wo.T + bo
// Fully fused per 64-token block. Weight k-panels staged in LDS (panel at LDS
// offset 0 so B-fragment ds_loads use immediate offsets, no per-iter VALU).

#define E_DIM 256
#define TOK   64          // tokens per block
#define XP    260         // activation LDS row stride: %4==0, %64==4 -> conflict-free A-frags
#define WP    272         // weight-panel LDS row stride: 2*WP%64==32 -> conflict-free B-frags
#define KP    32          // weight panel depth (k rows per stage)
#define NTOK_TOTAL (8 * 4096)

typedef __attribute__((ext_vector_type(2))) float v2f;
typedef __attribute__((ext_vector_type(4))) float v4f;
typedef __attribute__((ext_vector_type(8))) float v8f;

// ---------------------------------------------------------------------------
// One wave: 2 row-tiles (32 tokens) x 4 col-tiles (64 features), K=256.
// Weights staged through an LDS panel (at LDS offset 0) shared by the block.
// ---------------------------------------------------------------------------
template <bool TO_LDS>
__device__ __forceinline__ void gemm_tiles(const float* __restrict__ aLds,   // [TOK][XP]
                                           const float* __restrict__ wtg,    // global Wt: wtg[k*E+n]=W[n][k]
                                           const float* __restrict__ bias,   // [E]
                                           float*       __restrict__ dst,    // LDS [TOK][XP] or global (stride E)
                                           float*       __restrict__ wp,     // LDS panel [KP][WP] at offset 0
                                           int tid, int lane, int rowBase, int colBase)
{
    const int l15  = lane & 15;
    const int koff = (lane >> 4) << 1;                 // 0 or 2
    const float* arow0 = aLds + (rowBase + l15) * XP;
    const float* arow1 = arow0 + 16 * XP;
    // per-wave B base: all hot-loop offsets from here are compile-time consts
    const float* wbase = wp + koff * WP + colBase + l15;

    v8f acc[2][4];
    #pragma unroll
    for (int r = 0; r < 2; ++r)
        #pragma unroll
        for (int t = 0; t < 4; ++t) acc[r][t] = {};

    for (int kp = 0; kp < E_DIM; kp += KP) {
        __syncthreads();                               // previous panel fully consumed
        // ---- stage Wt[kp..kp+KP)[0..256) into LDS (32 KB, coalesced) ----
        for (int i = tid; i < KP * (E_DIM / 4); i += 256) {
            const int r = i >> 6;                      // 64 float4 per row
            const int c = (i & 63) << 2;
            const v4f v4 = *(const v4f*)(wtg + (size_t)(kp + r) * E_DIM + c);
            *(v4f*)(wp + r * WP + c) = v4;
        }
        // prefetch next panel into L2 while we compute on this one
        if (kp + KP < E_DIM) {
            const char* nxt = (const char*)(wtg + (size_t)(kp + KP) * E_DIM) + tid * 128;
            __builtin_prefetch(nxt, 0, 1);
        }
        __syncthreads();

        // ---- compute: 8 wmma per k-step, B-frags from LDS reused over 2 row-tiles ----
        #pragma unroll 2
        for (int kb = 0; kb < KP; kb += 4) {
            v2f a0, a1;
            a0.x = arow0[kp + kb + koff];
            a0.y = arow0[kp + kb + koff + 1];
            a1.x = arow1[kp + kb + koff];
            a1.y = arow1[kp + kb + koff + 1];
            #pragma unroll
            for (int t = 0; t < 4; ++t) {
                v2f b;
                b.x = wbase[kb * WP + t * 16];          // immediate DS offsets
                b.y = wbase[(kb + 1) * WP + t * 16];
                acc[0][t] = __builtin_amdgcn_wmma_f32_16x16x4_f32(
                    false, a0, false, b, (short)0, acc[0][t], false, false);
                acc[1][t] = __builtin_amdgcn_wmma_f32_16x16x4_f32(
                    false, a1, false, b, (short)0, acc[1][t], false, false);
            }
        }
    }

    // ---- epilogue: C/D layout VGPR v -> row v + 8*(lane>=16), col = lane&15 ----
    const int mB = rowBase + ((lane >> 4) << 3);
    const int dstStride = TO_LDS ? XP : E_DIM;
    #pragma unroll
    for (int r = 0; r < 2; ++r) {
        #pragma unroll
        for (int t = 0; t < 4; ++t) {
            const int n  = colBase + t * 16 + l15;
            const float bv = bias[n];
            #pragma unroll
            for (int v = 0; v < 8; ++v) {
                const float val = acc[r][t][v] + bv;
                if (TO_LDS) dst[(mB + r * 16 + v) * dstStride + n] = val;
                else __builtin_nontemporal_store(val, &dst[(mB + r * 16 + v) * dstStride + n]);
            }
        }
    }
}

// ---------------------------------------------------------------------------
// Weight pre-transpose: out[z][k][n] = W_z[n][k]   (4 x 256 x 256 f32 -> d_ws)
// ---------------------------------------------------------------------------
__global__ void transpose_w4(const float* __restrict__ w0, const float* __restrict__ w1,
                             const float* __restrict__ w2, const float* __restrict__ w3,
                             float* __restrict__ out)
{
    __shared__ float tile[16][17];
    const float* W = (blockIdx.z == 0) ? w0 : (blockIdx.z == 1) ? w1
                   : (blockIdx.z == 2) ? w2 : w3;
    float* o = out + (size_t)blockIdx.z * E_DIM * E_DIM;

    tile[threadIdx.y][threadIdx.x] =
        W[(blockIdx.y * 16 + threadIdx.y) * E_DIM + blockIdx.x * 16 + threadIdx.x];
    __syncthreads();
    o[(blockIdx.x * 16 + threadIdx.y) * E_DIM + (blockIdx.y * 16 + threadIdx.x)] =
        tile[threadIdx.x][threadIdx.y];
}

// ---------------------------------------------------------------------------
// Fused kernel: 256 threads = 8 wave32, 64 tokens per block.
// ---------------------------------------------------------------------------
__global__ void __launch_bounds__(256, 1)
attn_fused(const float* __restrict__ x,
           const float* __restrict__ wtq, const float* __restrict__ wtk,
           const float* __restrict__ wtv, const float* __restrict__ wto,
           const float* __restrict__ bq,  const float* __restrict__ bk,
           const float* __restrict__ bv,  const float* __restrict__ bo,
           float* __restrict__ out)
{
    __shared__ float smem[KP * WP + 4 * TOK * XP];    // ~301 KB of the 320 KB WGP LDS
    float* wp = smem;                                 // weight panel FIRST (offset 0)
    float* xs = wp + KP * WP;                         // x tile, later reused as y tile
    float* qs = xs + TOK * XP;
    float* ks = qs + TOK * XP;
    float* vs = ks + TOK * XP;

    const int tid     = threadIdx.x;
    const int lane    = tid & 31;
    const int wid     = tid >> 5;
    const int tokBase = blockIdx.x * TOK;

    // ---- stage x tile: 64 x 256 f32, coalesced non-temporal 128-bit loads ----
    for (int i = tid; i < TOK * (E_DIM / 4); i += 256) {
        const int r = i >> 6;
        const int c = (i & 63) << 2;
        const v4f v4 = __builtin_nontemporal_load(
            (const v4f*)(x + (size_t)(tokBase + r) * E_DIM + c));
        *(v4f*)(xs + r * XP + c) = v4;
    }

    const int rowBase = (wid >> 2) * 32;              // 0 or 32 (two 16-token tiles each)
    const int colBase = (wid & 3) * 64;               // 0,64,128,192

    // ---- QKV projections (panel loop syncs internally) ----
    gemm_tiles<true>(xs, wtq, bq, qs, wp, tid, lane, rowBase, colBase);
    gemm_tiles<true>(xs, wtk, bk, ks, wp, tid, lane, rowBase, colBase);
    gemm_tiles<true>(xs, wtv, bv, vs, wp, tid, lane, rowBase, colBase);
    __syncthreads();

    // ---- per-token attention over heads: scores are 4x4 (two 32-token passes) ----
    #pragma unroll
    for (int tp = 0; tp < 2; ++tp) {
        const int tok   = (tp << 5) + (tid >> 3);     // 8 lanes per token
        const int l8    = tid & 7;
        const int dbase = l8 * 8;

        float qf[4][8], kf[4][8], vf[4][8];
        #pragma unroll
        for (int h = 0; h < 4; ++h) {
            const float* qp  = qs + tok * XP + h * 64 + dbase;
            const float* kp2 = ks + tok * XP + h * 64 + dbase;
            const float* vp  = vs + tok * XP + h * 64 + dbase;
            #pragma unroll
            for (int j = 0; j < 8; ++j) { qf[h][j] = qp[j]; kf[h][j] = kp2[j]; vf[h][j] = vp[j]; }
        }

        float s[4][4];
        #pragma unroll
        for (int h = 0; h < 4; ++h)
            #pragma unroll
            for (int g = 0; g < 4; ++g) {
                float a = 0.f;
                #pragma unroll
                for (int j = 0; j < 8; ++j) a += qf[h][j] * kf[g][j];
                s[h][g] = a;
            }
        #pragma unroll
        for (int m = 1; m < 8; m <<= 1)
            #pragma unroll
            for (int h = 0; h < 4; ++h)
                #pragma unroll
                for (int g = 0; g < 4; ++g)
                    s[h][g] += __shfl_xor(s[h][g], m, 32);

        #pragma unroll
        for (int h = 0; h < 4; ++h) {
            float mx = s[h][0];
            #pragma unroll
            for (int g = 1; g < 4; ++g) mx = fmaxf(mx, s[h][g]);
            float p[4], sum = 0.f;
            #pragma unroll
            for (int g = 0; g < 4; ++g) { p[g] = __expf((s[h][g] - mx) * 0.125f); sum += p[g]; }
            const float inv = 1.0f / sum;
            #pragma unroll
            for (int g = 0; g < 4; ++g) p[g] *= inv;
            #pragma unroll
            for (int j = 0; j < 8; ++j) {
                float yv = 0.f;
                #pragma unroll
                for (int g = 0; g < 4; ++g) yv += p[g] * vf[g][j];
                xs[tok * XP + h * 64 + dbase + j] = yv;   // y overwrites x tile
            }
        }
    }
    __syncthreads();

    // ---- output projection, streamed straight to HBM (non-temporal) ----
    gemm_tiles<false>(xs, wto, bo, out + (size_t)tokBase * E_DIM, wp,
                      tid, lane, rowBase, colBase);
}

// ---------------------------------------------------------------------------
extern "C" void kernel_launch(void* const* d_in, const int* in_sizes, int n_in,
                              void* d_out, int out_size, void* d_ws, size_t ws_size,
                              hipStream_t stream)
{
    (void)in_sizes; (void)n_in; (void)out_size; (void)ws_size;
    const float* x  = (const float*)d_in[0];
    const float* Wq = (const float*)d_in[1];
    const float* bq = (const float*)d_in[2];
    const float* Wk = (const float*)d_in[3];
    const float* bk = (const float*)d_in[4];
    const float* Wv = (const float*)d_in[5];
    const float* bv = (const float*)d_in[6];
    const float* Wo = (const float*)d_in[7];
    const float* bo = (const float*)d_in[8];

    float* ws  = (float*)d_ws;                    // 4 * 256*256 floats = 1 MB
    float* wtq = ws;
    float* wtk = ws + 1 * E_DIM * E_DIM;
    float* wtv = ws + 2 * E_DIM * E_DIM;
    float* wto = ws + 3 * E_DIM * E_DIM;

    dim3 tb(16, 16), tg(E_DIM / 16, E_DIM / 16, 4);
    transpose_w4<<<tg, tb, 0, stream>>>(Wq, Wk, Wv, Wo, ws);

    attn_fused<<<NTOK_TOTAL / TOK, 256, 0, stream>>>(
        x, wtq, wtk, wtv, wto, bq, bk, bv, bo, (float*)d_out);
}